// S2Conv_28673201668340
// MI455X (gfx1250) — compile-verified
//
#include <hip/hip_runtime.h>
#include <hip/hip_bf16.h>

typedef __attribute__((ext_vector_type(2))) float v2f;
typedef __attribute__((ext_vector_type(8))) float v8f;

#define Nn 64
#define Cc 128
#define Tt 2048
#define Ss 9
#define TCHUNK 256
#define CHUNKS_PER_N (Tt / TCHUNK)     // 8
#define NBLOCKS (Nn * CHUNKS_PER_N)    // 512
#define EPSf 1e-5f

// Banded weight matrix entry: W[c][cc] = conv_w[c][cc-c+4] on the band, else 0.
// Out-of-range cc rows are all-zero, which neutralizes their B rows in WMMA.
__device__ __forceinline__ float wband(const float* __restrict__ w, int c, int cc) {
    int s = cc - c + (Ss / 2);
    if (cc < 0 || cc >= Cc || s < 0 || s >= Ss) return 0.f;
    return w[c * Ss + s];
}

// Per-wave fragment state shared by both conv kernels.
struct Frag {
    v2f a[6];                 // A-matrix fragments, 6 K-steps of 4
    const float* p0[6];       // fast path: xn + cc*Tt - sh + tb   (index by tt*16)
    const float* p1[6];
    int rb0[6], rb1[6];       // checked path: row base cc*Tt (clamped channel)
    int sh0[6], sh1[6];       // per-row time shift
};

__device__ __forceinline__ void setup_frags(Frag& f, const float* __restrict__ xn,
                                            const float* __restrict__ w,
                                            int c0, int half, int m, int tb) {
    const int crow = c0 + m;
#pragma unroll
    for (int kk = 0; kk < 6; ++kk) {
        int cck = c0 - (Ss / 2) + 4 * kk + 2 * half;
        int ccx = cck, ccy = cck + 1;
        f.a[kk].x = wband(w, crow, ccx);                 // 0 for invalid channels
        f.a[kk].y = wband(w, crow, ccy);
        int cx = ccx < 0 ? 0 : (ccx >= Cc ? Cc - 1 : ccx);
        int cy = ccy < 0 ? 0 : (ccy >= Cc ? Cc - 1 : ccy);
        int s0 = cx % Ss - (Ss / 2);
        int s1 = cy % Ss - (Ss / 2);
        f.sh0[kk] = s0;
        f.sh1[kk] = s1;
        f.rb0[kk] = cx * Tt;
        f.rb1[kk] = cy * Tt;
        f.p0[kk] = xn + (ptrdiff_t)(cx * Tt - s0 + tb);
        f.p1[kk] = xn + (ptrdiff_t)(cy * Tt - s1 + tb);
    }
}

// Interior tile: pure immediate-offset loads, then a 6-WMMA chain.
__device__ __forceinline__ v8f conv_tile_fast(const Frag& f, int ttoff) {
    v2f b[6];
#pragma unroll
    for (int kk = 0; kk < 6; ++kk) {
        b[kk].x = f.p0[kk][ttoff];
        b[kk].y = f.p1[kk][ttoff];
    }
    v8f acc = {};
#pragma unroll
    for (int kk = 0; kk < 6; ++kk) {
        acc = __builtin_amdgcn_wmma_f32_16x16x4_f32(
            false, f.a[kk], false, b[kk], (short)0, acc, false, false);
    }
    return acc;
}

// Edge tile: branchless clamp + cndmask zeroing for the time border.
__device__ __forceinline__ v8f conv_tile_checked(const Frag& f,
                                                 const float* __restrict__ xn, int t) {
    v2f b[6];
#pragma unroll
    for (int kk = 0; kk < 6; ++kk) {
        int t0i = t - f.sh0[kk];
        int t1i = t - f.sh1[kk];
        int t0c = t0i < 0 ? 0 : (t0i > Tt - 1 ? Tt - 1 : t0i);   // med3 clamp
        int t1c = t1i < 0 ? 0 : (t1i > Tt - 1 ? Tt - 1 : t1i);
        float vx = xn[f.rb0[kk] + t0c];                          // unconditional load
        float vy = xn[f.rb1[kk] + t1c];
        b[kk].x = (t0i == t0c) ? vx : 0.f;                       // cndmask, no branch
        b[kk].y = (t1i == t1c) ? vy : 0.f;
    }
    v8f acc = {};
#pragma unroll
    for (int kk = 0; kk < 6; ++kk) {
        acc = __builtin_amdgcn_wmma_f32_16x16x4_f32(
            false, f.a[kk], false, b[kk], (short)0, acc, false, false);
    }
    return acc;
}

// Pass 1: conv via WMMA; per-channel partial sum / sumsq per block (non-atomic).
__global__ __launch_bounds__(256) void conv_stats_kernel(
    const float* __restrict__ x, const float* __restrict__ w,
    float* __restrict__ psum, float* __restrict__ psumsq) {
    const int bid   = blockIdx.x;
    const int n     = bid / CHUNKS_PER_N;
    const int tbase = (bid % CHUNKS_PER_N) * TCHUNK;
    const int wave  = threadIdx.x >> 5;
    const int lane  = threadIdx.x & 31;
    const int half  = lane >> 4;
    const int m     = lane & 15;
    const int c0    = wave * 16;
    const int tb    = tbase + m;

    const float* xn = x + (size_t)n * Cc * Tt;
    Frag f;
    setup_frags(f, xn, w, c0, half, m, tb);

    v8f sumr = {}, sumq = {};
    if (tbase != 0 && tbase != Tt - TCHUNK) {
#pragma unroll
        for (int tt = 0; tt < TCHUNK / 16; ++tt) {
            v8f acc = conv_tile_fast(f, tt * 16);
#pragma unroll
            for (int v = 0; v < 8; ++v) {
                sumr[v] += acc[v];
                sumq[v] += acc[v] * acc[v];
            }
        }
    } else {
#pragma unroll 2
        for (int tt = 0; tt < TCHUNK / 16; ++tt) {
            v8f acc = conv_tile_checked(f, xn, tb + tt * 16);
#pragma unroll
            for (int v = 0; v < 8; ++v) {
                sumr[v] += acc[v];
                sumq[v] += acc[v] * acc[v];
            }
        }
    }

    // Reduce across each 16-lane half (a row's 16 t-values live across lanes).
#pragma unroll
    for (int v = 0; v < 8; ++v) {
        float s = sumr[v], q = sumq[v];
#pragma unroll
        for (int off = 1; off < 16; off <<= 1) {
            s += __shfl_xor(s, off, 16);
            q += __shfl_xor(q, off, 16);
        }
        if (m == 0) {
            int c = c0 + v + 8 * half;
            psum[bid * Cc + c]   = s;
            psumsq[bid * Cc + c] = q;
        }
    }
}

// Pass 2: fold partials + gamma/beta into per-channel scale/bias.
__global__ __launch_bounds__(128) void stats_finalize_kernel(
    const float* __restrict__ psum, const float* __restrict__ psumsq,
    const float* __restrict__ gamma, const float* __restrict__ beta,
    float* __restrict__ scale, float* __restrict__ bias) {
    const int c = threadIdx.x;
    float s = 0.f, q = 0.f;
#pragma unroll 4
    for (int b = 0; b < NBLOCKS; ++b) {
        s += psum[b * Cc + c];
        q += psumsq[b * Cc + c];
    }
    const float inv_nt = 1.0f / (float)(Nn * Tt);
    float mean = s * inv_nt;
    float var  = q * inv_nt - mean * mean;
    float rstd = rsqrtf(var + EPSf);
    float g = gamma[c];
    scale[c] = rstd * g;
    bias[c]  = beta[c] - mean * rstd * g;
}

// Pass 3: recompute conv via WMMA (x is L2-resident), normalize + ReLU, store.
__global__ __launch_bounds__(256) void conv_norm_kernel(
    const float* __restrict__ x, const float* __restrict__ w,
    const float* __restrict__ scale, const float* __restrict__ bias,
    float* __restrict__ out) {
    const int bid   = blockIdx.x;
    const int n     = bid / CHUNKS_PER_N;
    const int tbase = (bid % CHUNKS_PER_N) * TCHUNK;
    const int wave  = threadIdx.x >> 5;
    const int lane  = threadIdx.x & 31;
    const int half  = lane >> 4;
    const int m     = lane & 15;
    const int c0    = wave * 16;
    const int tb    = tbase + m;

    const float* xn = x + (size_t)n * Cc * Tt;
    Frag f;
    setup_frags(f, xn, w, c0, half, m, tb);

    float sc[8], bi[8];
    float* po[8];
    float* outn = out + (size_t)n * Cc * Tt;
#pragma unroll
    for (int v = 0; v < 8; ++v) {
        int c = c0 + v + 8 * half;
        sc[v] = scale[c];
        bi[v] = bias[c];
        po[v] = outn + (size_t)c * Tt + tb;   // store with immediate offsets
    }

    if (tbase != 0 && tbase != Tt - TCHUNK) {
#pragma unroll
        for (int tt = 0; tt < TCHUNK / 16; ++tt) {
            v8f acc = conv_tile_fast(f, tt * 16);
#pragma unroll
            for (int v = 0; v < 8; ++v) {
                float r = acc[v] * sc[v] + bi[v];
                po[v][tt * 16] = fmaxf(r, 0.f);
            }
        }
    } else {
#pragma unroll 2
        for (int tt = 0; tt < TCHUNK / 16; ++tt) {
            v8f acc = conv_tile_checked(f, xn, tb + tt * 16);
#pragma unroll
            for (int v = 0; v < 8; ++v) {
                float r = acc[v] * sc[v] + bi[v];
                po[v][tt * 16] = fmaxf(r, 0.f);
            }
        }
    }
}

extern "C" void kernel_launch(void* const* d_in, const int* in_sizes, int n_in,
                              void* d_out, int out_size, void* d_ws, size_t ws_size,
                              hipStream_t stream) {
    const float* x     = (const float*)d_in[0];
    const float* w     = (const float*)d_in[1];
    const float* gamma = (const float*)d_in[2];
    const float* beta  = (const float*)d_in[3];
    float* out = (float*)d_out;

    float* psum   = (float*)d_ws;              // NBLOCKS*Cc
    float* psumsq = psum + NBLOCKS * Cc;       // NBLOCKS*Cc
    float* scale  = psumsq + NBLOCKS * Cc;     // Cc
    float* bias   = scale + Cc;                // Cc

    conv_stats_kernel<<<NBLOCKS, 256, 0, stream>>>(x, w, psum, psumsq);
    stats_finalize_kernel<<<1, 128, 0, stream>>>(psum, psumsq, gamma, beta, scale, bias);
    conv_norm_kernel<<<NBLOCKS, 256, 0, stream>>>(x, w, scale, bias, out);
}